// CapsuleLayer__75634374082758
// MI455X (gfx1250) — compile-verified
//
#include <hip/hip_runtime.h>

#define B_ 64
#define I_ 512
#define D_ 128
#define O_ 32
#define H_ 64

typedef __attribute__((ext_vector_type(2))) float v2f;
typedef __attribute__((ext_vector_type(8))) float v8f;

// ---------------------------------------------------------------------------
// Kernel A: xs[b][d] = sum_i x[b][i][d]           (64 blocks x 128 threads)
// ---------------------------------------------------------------------------
__global__ void k_colsum(const float* __restrict__ x, float* __restrict__ xs) {
  int b = blockIdx.x;
  int d = threadIdx.x;  // 0..127, coalesced across d
  const float* xp = x + (size_t)b * I_ * D_ + d;
  float s = 0.f;
#pragma unroll 8
  for (int i = 0; i < I_; ++i) s += xp[(size_t)i * D_];
  xs[b * D_ + d] = s;
}

// ---------------------------------------------------------------------------
// Kernel B: iter-0.  v0[h] = (1/O)*sum_d xs[b,d]*W[o,d,h]; out0 = squash(v0);
//           u[b,o,d] = sum_h W[o,d,h]*out0[h]      (B*O blocks x 64 threads)
// ---------------------------------------------------------------------------
__global__ void k_iter0(const float* __restrict__ xs, const float* __restrict__ W,
                        float* __restrict__ u) {
  int bo = blockIdx.x;
  int b = bo / O_, o = bo % O_;
  int h = threadIdx.x;  // 0..63
  __shared__ float out0[H_];
  __shared__ float red[H_];

  const float* Wo  = W + (size_t)o * D_ * H_;
  const float* xsb = xs + b * D_;

  float v = 0.f;
#pragma unroll 4
  for (int d = 0; d < D_; ++d) v += xsb[d] * Wo[d * H_ + h];
  v *= (1.0f / O_);

  red[h] = v * v;
  __syncthreads();
  if (h == 0) {
    float sn = 0.f;
    for (int t = 0; t < H_; ++t) sn += red[t];
    red[0] = sn;
  }
  __syncthreads();
  float sn = red[0];
  float scale = (sn / (1.f + sn)) * rsqrtf(sn);
  out0[h] = v * scale;
  __syncthreads();

  float* ub = u + (size_t)bo * D_;
  for (int dd = h; dd < D_; dd += H_) {
    const float* Wd = Wo + dd * H_;
    float acc = 0.f;
#pragma unroll 4
    for (int hh = 0; hh < H_; ++hh) acc += Wd[hh] * out0[hh];
    ub[dd] = acc;
  }
}

// ---------------------------------------------------------------------------
// Kernel C: logits[b,i,o] = sum_d x[b,i,d]*u[b,o,d], softmax over o,
//           store probs[b][o][i].   WMMA fp32 16x16x4, M=i N=o K=d.
//           Grid: B*(I/16) blocks of 1 wave (32 threads).
// ---------------------------------------------------------------------------
__global__ void k_logits_softmax(const float* __restrict__ x,
                                 const float* __restrict__ u,
                                 float* __restrict__ probs) {
  int blk = blockIdx.x;
  int b  = blk >> 5;            // I/16 == 32 tiles per batch
  int i0 = (blk & 31) * 16;
  int l  = threadIdx.x;         // 0..31
  int lh = l & 15;
  int kh = (l >> 4) * 2;        // K sub-offset: lanes 0-15 -> {0,1}, 16-31 -> {2,3}

  const float* xb = x + (size_t)b * I_ * D_;
  const float* ub = u + (size_t)b * O_ * D_;

  // A row (M = i0+lh), B rows (N = o = lh / lh+16); float2 covers the K pair.
  const float* arow  = xb + (size_t)(i0 + lh) * D_ + kh;
  const float* brow0 = ub + (size_t)lh * D_ + kh;
  const float* brow1 = ub + (size_t)(lh + 16) * D_ + kh;

  v8f acc0 = {};
  v8f acc1 = {};
#pragma unroll 4
  for (int k = 0; k < D_; k += 4) {
    v2f a  = *(const v2f*)(arow + k);
    v2f b0 = *(const v2f*)(brow0 + k);
    v2f b1 = *(const v2f*)(brow1 + k);
    acc0 = __builtin_amdgcn_wmma_f32_16x16x4_f32(false, a, false, b0, (short)0,
                                                 acc0, false, false);
    acc1 = __builtin_amdgcn_wmma_f32_16x16x4_f32(false, a, false, b1, (short)0,
                                                 acc1, false, false);
  }

  // C layout: vgpr r, lanes 0-15: (i = i0+r,   o = lh)      [acc1: o = lh+16]
  //                   lanes 16-31:(i = i0+8+r, o = lh)      [acc1: o = lh+16]
  // Softmax over o for a fixed row lives in the 16 lanes of this half-wave
  // across acc0 & acc1 -> xor-shuffle reduction with masks {8,4,2,1}.
  float pr0[8], pr1[8];
#pragma unroll
  for (int r = 0; r < 8; ++r) {
    float m = fmaxf(acc0[r], acc1[r]);
    m = fmaxf(m, __shfl_xor(m, 8, 32));
    m = fmaxf(m, __shfl_xor(m, 4, 32));
    m = fmaxf(m, __shfl_xor(m, 2, 32));
    m = fmaxf(m, __shfl_xor(m, 1, 32));
    float e0 = __expf(acc0[r] - m);
    float e1 = __expf(acc1[r] - m);
    float s = e0 + e1;
    s += __shfl_xor(s, 8, 32);
    s += __shfl_xor(s, 4, 32);
    s += __shfl_xor(s, 2, 32);
    s += __shfl_xor(s, 1, 32);
    float inv = 1.0f / s;
    pr0[r] = e0 * inv;
    pr1[r] = e1 * inv;
  }

  float* pb = probs + (size_t)b * O_ * I_;
  int irow = i0 + 8 * (l >> 4);
#pragma unroll
  for (int r = 0; r < 8; ++r) {
    pb[(size_t)lh * I_ + irow + r]        = pr0[r];
    pb[(size_t)(lh + 16) * I_ + irow + r] = pr1[r];
  }
}

// ---------------------------------------------------------------------------
// Kernel D: y[b,o,d] = sum_i probs[b,o,i]*x[b,i,d].  WMMA fp32 16x16x4,
//           M=o N=d K=i.  Grid: B*2*8 blocks of 1 wave.
// ---------------------------------------------------------------------------
__global__ void k_y(const float* __restrict__ probs, const float* __restrict__ x,
                    float* __restrict__ y) {
  int blk = blockIdx.x;
  int b  = blk >> 4;
  int t  = blk & 15;
  int m0 = (t >> 3) * 16;  // o tile: 0 or 16
  int n0 = (t & 7) * 16;   // d tile: 0..112
  int l  = threadIdx.x;
  int lh = l & 15;
  int kh = (l >> 4) * 2;

  const float* pa = probs + (size_t)b * O_ * I_ + (size_t)(m0 + lh) * I_ + kh;
  const float* xb = x + (size_t)b * I_ * D_ + n0 + lh;

  v8f acc = {};
#pragma unroll 4
  for (int k = 0; k < I_; k += 4) {
    v2f a = *(const v2f*)(pa + k);          // K pair along i, contiguous
    v2f bb;                                  // B[k', n]: column of x, stride D_
    bb.x = xb[(size_t)(k + kh) * D_];
    bb.y = xb[(size_t)(k + kh + 1) * D_];
    acc = __builtin_amdgcn_wmma_f32_16x16x4_f32(false, a, false, bb, (short)0,
                                                acc, false, false);
  }

  // C layout: vgpr r, lanes 0-15: (o = m0+r, d = n0+lh); lanes 16-31: o = m0+8+r
  float* yb = y + (size_t)b * O_ * D_;
  int orow = m0 + 8 * (l >> 4);
#pragma unroll
  for (int r = 0; r < 8; ++r)
    yb[(size_t)(orow + r) * D_ + n0 + lh] = acc[r];
}

// ---------------------------------------------------------------------------
// Kernel E: v1[b,o,h] = sum_d y[b,o,d]*W[o,d,h]; out = squash(v1).
//           (B*O blocks x 64 threads)
// ---------------------------------------------------------------------------
__global__ void k_out(const float* __restrict__ y, const float* __restrict__ W,
                      float* __restrict__ out) {
  int bo = blockIdx.x;
  int o = bo % O_;
  int h = threadIdx.x;  // 0..63
  __shared__ float red[H_];

  const float* yb = y + (size_t)bo * D_;
  const float* Wo = W + (size_t)o * D_ * H_;

  float v = 0.f;
#pragma unroll 4
  for (int d = 0; d < D_; ++d) v += yb[d] * Wo[d * H_ + h];

  red[h] = v * v;
  __syncthreads();
  if (h == 0) {
    float sn = 0.f;
    for (int t = 0; t < H_; ++t) sn += red[t];
    red[0] = sn;
  }
  __syncthreads();
  float sn = red[0];
  float scale = (sn / (1.f + sn)) * rsqrtf(sn);
  out[(size_t)bo * H_ + h] = v * scale;
}

// ---------------------------------------------------------------------------
extern "C" void kernel_launch(void* const* d_in, const int* in_sizes, int n_in,
                              void* d_out, int out_size, void* d_ws, size_t ws_size,
                              hipStream_t stream) {
  const float* x = (const float*)d_in[0];        // [B, I, D]
  const float* W = (const float*)d_in[1];        // [O, D, H]
  float* out = (float*)d_out;                    // [B, O, H]

  char* ws = (char*)d_ws;
  float* xs = (float*)ws;    ws += (size_t)B_ * D_ * sizeof(float);        // 32 KB
  float* u  = (float*)ws;    ws += (size_t)B_ * O_ * D_ * sizeof(float);   // 1 MB
  float* pr = (float*)ws;    ws += (size_t)B_ * O_ * I_ * sizeof(float);   // 4 MB
  float* y  = (float*)ws;                                                  // 1 MB

  k_colsum<<<B_, D_, 0, stream>>>(x, xs);
  k_iter0<<<B_ * O_, H_, 0, stream>>>(xs, W, u);
  k_logits_softmax<<<B_ * (I_ / 16), 32, 0, stream>>>(x, u, pr);
  k_y<<<B_ * 16, 32, 0, stream>>>(pr, x, y);
  k_out<<<B_ * O_, H_, 0, stream>>>(y, W, out);
}